// DxPrivacy_34325378630099
// MI455X (gfx1250) — compile-verified
//
#include <hip/hip_runtime.h>
#include <math.h>

typedef __attribute__((ext_vector_type(2))) float v2f;
typedef __attribute__((ext_vector_type(8))) float v8f;

#define VOCAB        50257
#define DMODEL       768
#define NROWS        4096            // 2 * 2048
#define ROW_TILES    256             // NROWS / 16
#define CHUNK_COLS   3904            // 244 tiles of 16 columns
#define N_CHUNKS     13              // ceil(50257 / 3904)
#define TILES_PER_CHUNK (CHUNK_COLS / 16)
#define LDS_STRIDE   772             // 768 + 4 pad: kills 64-bank conflicts, keeps 16B align

// ---------------------------------------------------------------------------
// Kernel 1: Laplace noise via JAX Threefry-2x32 (key(42) -> (0,42)), add to x.
// Element i < half uses x0-output of pair (i, i+half); i >= half uses x1.
// ---------------------------------------------------------------------------
__device__ __forceinline__ unsigned rotl32(unsigned x, unsigned n) {
    return (x << n) | (x >> (32u - n));
}

__device__ __forceinline__ void threefry2x32_0_42(unsigned x0, unsigned x1,
                                                  unsigned& o0, unsigned& o1) {
    const unsigned k0 = 0u, k1 = 42u, k2 = 0x1BD11BDAu ^ k0 ^ k1;
    x0 += k0; x1 += k1;
#define TFR(r) { x0 += x1; x1 = rotl32(x1, r); x1 ^= x0; }
    TFR(13u) TFR(15u) TFR(26u) TFR(6u)   x0 += k1; x1 += k2 + 1u;
    TFR(17u) TFR(29u) TFR(16u) TFR(24u)  x0 += k2; x1 += k0 + 2u;
    TFR(13u) TFR(15u) TFR(26u) TFR(6u)   x0 += k0; x1 += k1 + 3u;
    TFR(17u) TFR(29u) TFR(16u) TFR(24u)  x0 += k1; x1 += k2 + 4u;
    TFR(13u) TFR(15u) TFR(26u) TFR(6u)   x0 += k2; x1 += k0 + 5u;
#undef TFR
    o0 = x0; o1 = x1;
}

__device__ __forceinline__ float laplace_from_bits(unsigned bits) {
    // JAX uniform(minval=-1+2^-24, maxval=1): f in [0,1) from mantissa bits
    const float minv = -1.0f + 5.9604645e-08f;     // -1 + 2^-24
    float f = __uint_as_float((bits >> 9) | 0x3f800000u) - 1.0f;
    float u = fmaf(f, 1.0f - minv, minv);
    u = fmaxf(minv, u);
    float s = (u > 0.0f) ? 1.0f : ((u < 0.0f) ? -1.0f : 0.0f);
    float lap = s * log1pf(-fabsf(u));             // JAX: sign(u)*log1p(-|u|)
    return lap / 5.0f;                             // scale 1/EPSILON
}

__global__ __launch_bounds__(256) void noise_kernel(const float* __restrict__ x,
                                                    float* __restrict__ noisy,
                                                    int half) {
    int t = blockIdx.x * blockDim.x + threadIdx.x;
    if (t >= half) return;
    unsigned o0, o1;
    threefry2x32_0_42((unsigned)t, (unsigned)(t + half), o0, o1);
    noisy[t]        = x[t]        + laplace_from_bits(o0);
    noisy[t + half] = x[t + half] + laplace_from_bits(o1);
}

// ---------------------------------------------------------------------------
// Kernel 2: fused fp32-WMMA GEMM + per-row running argmax over a vocab chunk.
// Block = 256 threads = 8 waves; one 16-row tile per block; waves stride over
// the chunk's 16-column tiles. A (16x768 f32) staged in LDS once per block.
// ---------------------------------------------------------------------------
__global__ __launch_bounds__(256) void simargmax_kernel(const float* __restrict__ noisy,
                                                        const float* __restrict__ E,
                                                        float* __restrict__ pval,
                                                        int* __restrict__ pidx) {
    __shared__ float Atile[16 * LDS_STRIDE];
    __shared__ float wvals[8][16];
    __shared__ int   widxs[8][16];

    const int tid     = threadIdx.x;
    const int rowTile = blockIdx.x;
    const int chunk   = blockIdx.y;

    // Cooperative float4 load of the A tile (16 x 768) into padded LDS.
    const float4* src = (const float4*)(noisy + (size_t)rowTile * 16 * DMODEL);
    for (int i = tid; i < 16 * (DMODEL / 4); i += 256) {
        int r  = i / (DMODEL / 4);
        int k4 = i % (DMODEL / 4);
        *(float4*)&Atile[r * LDS_STRIDE + k4 * 4] = src[r * (DMODEL / 4) + k4];
    }
    __syncthreads();

    const int wave   = tid >> 5;
    const int lane   = tid & 31;
    const int l15    = lane & 15;
    const int koff   = (lane >> 4) * 2;   // lanes 0-15: K{0,1}; lanes 16-31: K{2,3}
    const float* Abase = &Atile[l15 * LDS_STRIDE + koff];

    float bestv[8];
    int   besti[8];
#pragma unroll
    for (int r = 0; r < 8; ++r) { bestv[r] = -INFINITY; besti[r] = 0x7fffffff; }

    const int chunkBase = chunk * CHUNK_COLS;
    for (int t = wave; t < TILES_PER_CHUNK; t += 8) {
        const int v0    = chunkBase + t * 16;
        const int v     = v0 + l15;                  // this lane's vocab column
        const bool valid = (v < VOCAB);
        const int vc    = valid ? v : (VOCAB - 1);   // clamp: safe load, masked later
        const float* Brow = E + (size_t)vc * DMODEL + koff;

        v8f acc = {};
#pragma unroll 4
        for (int k = 0; k < DMODEL; k += 4) {
            v2f a = *(const v2f*)(Abase + k);        // ds_load_b64
            v2f b = *(const v2f*)(Brow + k);         // global_load_b64 (L2-resident)
            acc = __builtin_amdgcn_wmma_f32_16x16x4_f32(
                false, a, false, b, (short)0, acc, false, false);
        }

#pragma unroll
        for (int r = 0; r < 8; ++r) {
            float val = acc[r];
            if (valid && val > bestv[r]) { bestv[r] = val; besti[r] = v; }
        }
    }

    // Cross-lane reduce within 16-lane halves (C/D layout: VGPR r -> row r / r+8).
#pragma unroll
    for (int m = 1; m <= 8; m <<= 1) {
#pragma unroll
        for (int r = 0; r < 8; ++r) {
            float ov = __shfl_xor(bestv[r], m, 32);
            int   oi = __shfl_xor(besti[r], m, 32);
            if (ov > bestv[r] || (ov == bestv[r] && oi < besti[r])) {
                bestv[r] = ov; besti[r] = oi;
            }
        }
    }

    if (lane == 0) {
#pragma unroll
        for (int r = 0; r < 8; ++r) { wvals[wave][r] = bestv[r]; widxs[wave][r] = besti[r]; }
    }
    if (lane == 16) {
#pragma unroll
        for (int r = 0; r < 8; ++r) { wvals[wave][r + 8] = bestv[r]; widxs[wave][r + 8] = besti[r]; }
    }
    __syncthreads();

    if (tid < 16) {
        float bv = wvals[0][tid];
        int   bi = widxs[0][tid];
#pragma unroll
        for (int w = 1; w < 8; ++w) {
            float ov = wvals[w][tid];
            int   oi = widxs[w][tid];
            if (ov > bv || (ov == bv && oi < bi)) { bv = ov; bi = oi; }
        }
        const int row = rowTile * 16 + tid;
        pval[chunk * NROWS + row] = bv;
        pidx[chunk * NROWS + row] = bi;
    }
}

// ---------------------------------------------------------------------------
// Kernel 3: reduce chunk partials to the final argmax index per row.
// ---------------------------------------------------------------------------
__global__ __launch_bounds__(256) void reduce_kernel(const float* __restrict__ pval,
                                                     const int* __restrict__ pidx,
                                                     int* __restrict__ fidx) {
    int row = blockIdx.x * blockDim.x + threadIdx.x;
    if (row >= NROWS) return;
    float bv = pval[row];
    int   bi = pidx[row];
    for (int c = 1; c < N_CHUNKS; ++c) {
        float v = pval[c * NROWS + row];
        int   i = pidx[c * NROWS + row];
        if (v > bv || (v == bv && i < bi)) { bv = v; bi = i; }
    }
    fidx[row] = bi;
}

// ---------------------------------------------------------------------------
// Kernel 4: gather winning embedding rows, float4-vectorized.
// ---------------------------------------------------------------------------
__global__ __launch_bounds__(256) void gather_kernel(const float* __restrict__ E,
                                                     const int* __restrict__ fidx,
                                                     float* __restrict__ out) {
    int g   = blockIdx.x * blockDim.x + threadIdx.x;  // NROWS * 192 threads exactly
    int row = g / (DMODEL / 4);
    int j   = g % (DMODEL / 4);
    const float4* Erow = (const float4*)(E + (size_t)fidx[row] * DMODEL);
    ((float4*)out)[(size_t)row * (DMODEL / 4) + j] = Erow[j];
}

// ---------------------------------------------------------------------------
extern "C" void kernel_launch(void* const* d_in, const int* in_sizes, int n_in,
                              void* d_out, int out_size, void* d_ws, size_t ws_size,
                              hipStream_t stream) {
    (void)in_sizes; (void)n_in; (void)out_size; (void)ws_size;
    const float* x = (const float*)d_in[0];   // inputs_embeds [2,2048,768] f32
    const float* E = (const float*)d_in[1];   // embed_table  [50257,768]  f32
    float* out = (float*)d_out;               // [2,2048,768] f32

    char* ws = (char*)d_ws;
    float* noisy = (float*)ws;
    size_t off = (size_t)NROWS * DMODEL * sizeof(float);           // 12.58 MB
    float* pval = (float*)(ws + off); off += (size_t)N_CHUNKS * NROWS * sizeof(float);
    int*   pidx = (int*)(ws + off);   off += (size_t)N_CHUNKS * NROWS * sizeof(int);
    int*   fidx = (int*)(ws + off);

    const int half = NROWS * DMODEL / 2;
    noise_kernel<<<(half + 255) / 256, 256, 0, stream>>>(x, noisy, half);

    dim3 grid(ROW_TILES, N_CHUNKS);
    simargmax_kernel<<<grid, 256, 0, stream>>>(noisy, E, pval, pidx);

    reduce_kernel<<<(NROWS + 255) / 256, 256, 0, stream>>>(pval, pidx, fidx);

    gather_kernel<<<(NROWS * (DMODEL / 4)) / 256, 256, 0, stream>>>(E, fidx, out);
}